// ranking_loss_14851996910048
// MI455X (gfx1250) — compile-verified
//
#include <hip/hip_runtime.h>
#include <stdint.h>

// ---------------------------------------------------------------------------
// Triplet margin loss, MI455X (gfx1250).
// Memory-bound: 480 MB negatives @ 23.3 TB/s => ~21 us floor.
// Strategy: async B128 global->LDS streaming (ASYNCcnt), squared row norms
// via V_WMMA_F32_16X16X4_F32 diagonal trick (D = X*X^T accumulate), scalar
// epilogue with wave reduce + one f32 atomic per wave.
// ---------------------------------------------------------------------------

#define DIMS        2400
#define NROWS       50000
#define NPOS        5
#define EPSV        1e-6f
#define MARGIN      1.0f

#define KC          96            // K-chunk per LDS tile (2400 = 25 * 96)
#define NTILES      (DIMS / KC)   // 25
#define ROWSTRIDE   100           // padded floats per LDS tile row (bank-conflict free)
#define ROWS_PER_WAVE 16
#define WAVES_PER_WG  4
#define WG_THREADS  (WAVES_PER_WG * 32)
#define TOTAL_WAVES (NROWS / ROWS_PER_WAVE)                 // 3125
#define NUM_WGS     ((TOTAL_WAVES + WAVES_PER_WG - 1) / WAVES_PER_WG) // 782

typedef float v2f __attribute__((ext_vector_type(2)));
typedef float v8f __attribute__((ext_vector_type(8)));

// --- kernel 1: d_ap[p] = ||a - pos_p + eps||, and zero the output scalar ----
__global__ void __launch_bounds__(192)
dap_kernel(const float* __restrict__ anchor,
           const float* __restrict__ pos,
           float* __restrict__ dap,
           float* __restrict__ out) {
    const int tid  = threadIdx.x;
    const int lane = tid & 31;
    const int w    = tid >> 5;           // wave id: 0..5
    if (tid == 0) *out = 0.0f;           // harness poisons d_out; re-zero every call
    if (w >= NPOS) return;
    float s = 0.0f;
    for (int k = lane; k < DIMS; k += 32) {
        float x = anchor[k] + EPSV - pos[w * DIMS + k];
        s = fmaf(x, x, s);
    }
    #pragma unroll
    for (int o = 16; o > 0; o >>= 1) s += __shfl_xor(s, o, 32);
    if (lane == 0) dap[w] = sqrtf(s);
}

// --- kernel 2: main streaming + WMMA reduction ------------------------------
__global__ void __launch_bounds__(WG_THREADS)
triplet_main(const float* __restrict__ anchor,
             const float* __restrict__ dap,
             const float* __restrict__ neg,
             float* __restrict__ out) {
    __shared__ __align__(16) float s_ae[DIMS];   // anchor + eps (broadcast reads)
    __shared__ __align__(16) float s_tile[WAVES_PER_WG][ROWS_PER_WAVE * ROWSTRIDE];

    const int tid  = threadIdx.x;
    const int lane = tid & 31;
    const int wave = tid >> 5;

    // Stage anchor(+eps) once per workgroup.
    for (int i = tid; i < DIMS; i += WG_THREADS) s_ae[i] = anchor[i] + EPSV;
    __syncthreads();

    const int waveGlobal = blockIdx.x * WAVES_PER_WG + wave;
    const int baseRow    = waveGlobal * ROWS_PER_WAVE;
    if (baseRow >= NROWS) return;        // wave-uniform; EXEC stays all-ones below

    float* tile = &s_tile[wave][0];
    const unsigned ldsBase = (unsigned)(uintptr_t)tile; // low 32 bits = LDS offset

    // WMMA A-matrix (16x4 f32) lane mapping: lanes 0-15 -> K={0,1},
    // lanes 16-31 -> K={2,3}; row M = lane & 15. B = A^T has the identical
    // register image, so wmma(x, x, acc) accumulates row norms on the diagonal.
    const int r    = lane & 15;
    const int koff = (lane >> 4) << 1;   // 0 or 2

    v8f acc = {};

    for (int t = 0; t < NTILES; ++t) {
        // ---- async global -> LDS: 16 rows x 96 floats = 384 x b128, 12/lane
        const float* gbase = neg + (size_t)baseRow * DIMS + t * KC;
        #pragma unroll
        for (int i = 0; i < 12; ++i) {
            int f   = i * 32 + lane;      // 0..383
            int row = f / 24;             // 24 float4 per row
            int c4  = f - row * 24;
            uint64_t gaddr  = (uint64_t)(uintptr_t)(gbase + (size_t)row * DIMS + c4 * 4);
            unsigned ldsOff = ldsBase + (unsigned)(row * ROWSTRIDE + c4 * 4) * 4u;
            asm volatile("global_load_async_to_lds_b128 %0, %1, off"
                         :: "v"(ldsOff), "v"(gaddr) : "memory");
        }
        asm volatile("s_wait_asynccnt 0" ::: "memory");

        // ---- 24 WMMA steps over this tile (K advances by 4 each step)
        #pragma unroll
        for (int k = 0; k < KC; k += 4) {
            float2 tv = *(const float2*)(tile + r * ROWSTRIDE + k + koff); // ds_load_b64
            float2 av = *(const float2*)(s_ae + t * KC + k + koff);        // broadcast
            v2f x;
            x.x = av.x - tv.x;           // (a + eps) - n
            x.y = av.y - tv.y;
            acc = __builtin_amdgcn_wmma_f32_16x16x4_f32(
                      false, x, false, x, (short)0, acc, false, false);
        }
    }

    // ---- diagonal extraction: lanes 0-7 hold acc[lane] (M=lane),
    //      lanes 24-31 hold acc[lane-24] (M=lane-16). Others inactive.
    int  sel    = (lane < 8) ? lane : (lane - 24);
    bool active = (lane < 8) || (lane >= 24);
    float diag = acc[0];
    #pragma unroll
    for (int j = 1; j < 8; ++j) diag = (sel == j) ? acc[j] : diag;

    int   row  = baseRow + ((lane < 8) ? lane : (lane - 16));
    float d_an = sqrtf(diag);
    float dp   = dap[row % NPOS];
    float c    = dp + MARGIN - d_an;
    c = (c > 0.0f) ? c : 0.0f;
    float contrib = active ? c : 0.0f;

    #pragma unroll
    for (int o = 16; o > 0; o >>= 1) contrib += __shfl_xor(contrib, o, 32);
    if (lane == 0) atomicAdd(out, contrib);   // global_atomic_add_f32
}

extern "C" void kernel_launch(void* const* d_in, const int* in_sizes, int n_in,
                              void* d_out, int out_size, void* d_ws, size_t ws_size,
                              hipStream_t stream) {
    const float* anchor = (const float*)d_in[0];   // (1, 2400)
    const float* pos    = (const float*)d_in[1];   // (5, 2400)
    const float* neg    = (const float*)d_in[2];   // (50000, 2400)
    float* out = (float*)d_out;                    // scalar loss
    float* dap = (float*)d_ws;                     // 5 floats scratch

    dap_kernel<<<1, 192, 0, stream>>>(anchor, pos, dap, out);
    triplet_main<<<NUM_WGS, WG_THREADS, 0, stream>>>(anchor, dap, neg, out);
}